// MultiARM_Sync_43112881717602
// MI455X (gfx1250) — compile-verified
//
#include <hip/hip_runtime.h>
#include <cmath>

typedef __attribute__((ext_vector_type(2))) float v2f;
typedef __attribute__((ext_vector_type(4))) float v4f;
typedef __attribute__((ext_vector_type(8))) float v8f;

// Problem constants: B=8, C=256, H=W=512, SPLIT=4 -> 128x128 tiles,
// 16 tiles/image, M = B*16 = 128 "rows" for the gate GEMM, K=N=256.

// ---------------------------------------------------------------------------
// Pass 1: per-(b,c,tile) mean over a 128x128 tile.
// One 256-thread block per (b,c,tile): 32768 blocks. Each wave reads one full
// 512B tile row per step (perfectly coalesced float4), LDS tree-reduce.
// Non-temporal loads: this 2 GiB stream is re-read only much later (pass 3),
// so don't let it occupy L2.
// pooled layout: pooled[(b*16 + tile)*256 + c]  (GEMM A matrix, row-major)
// ---------------------------------------------------------------------------
__global__ __launch_bounds__(256) void pool_kernel(const float* __restrict__ x,
                                                   float* __restrict__ pooled) {
  const int bid = blockIdx.x;          // b*4096 + c*16 + t
  const int t   = bid & 15;
  const int c   = (bid >> 4) & 255;
  const int b   = bid >> 12;
  const int ti  = t >> 2, tj = t & 3;  // tile row/col

  const float* base = x + (((size_t)(b * 256 + c) * 512 + (size_t)ti * 128) * 512
                           + (size_t)tj * 128);
  const int tid = threadIdx.x;

  float sum = 0.0f;
#pragma unroll
  for (int s = 0; s < 16; ++s) {
    const int f    = tid + s * 256;    // float4 slot 0..4095 within tile
    const int row  = f >> 5;           // 32 float4 per 128-float tile row
    const int col4 = f & 31;
    const v4f* p = (const v4f*)(base + (size_t)row * 512 + (size_t)col4 * 4);
    v4f v = __builtin_nontemporal_load(p);
    sum += v.x + v.y + v.z + v.w;
  }

  __shared__ float red[256];
  red[tid] = sum;
  __syncthreads();
#pragma unroll
  for (int off = 128; off > 0; off >>= 1) {
    if (tid < off) red[tid] += red[tid + off];
    __syncthreads();
  }
  if (tid == 0) {
    const int m = b * 16 + t;
    pooled[(size_t)m * 256 + c] = red[0] * (1.0f / 16384.0f);
  }
}

// ---------------------------------------------------------------------------
// Pass 2: gate GEMM + bias + sigmoid via V_WMMA_F32_16X16X4_F32.
// att[m][o] = sigmoid( sum_c pooled[m][c] * W[o][c] + bias[o] )
// M=128, N=256, K=256 -> 8x16 = 128 output tiles of 16x16, one wave each.
// A frag 16x4 f32 (ISA 7.12.2): lane l holds A[l%16][k0 + (l/16)*2 .. +1].
// B frag 4x16: symmetric; B[k][n] = W[n][k] -> contiguous pair from W row n.
// C/D 16x16 f32: VGPR v, lane l -> M = v + 8*(l>=16), N = l%16.
// ---------------------------------------------------------------------------
__global__ __launch_bounds__(32) void gate_kernel(const float* __restrict__ pooled,
                                                  const float* __restrict__ Wm,
                                                  const float* __restrict__ bias,
                                                  float* __restrict__ att) {
  const int tileid = blockIdx.x;   // 0..127
  const int mt   = tileid >> 4;    // 0..7
  const int nt   = tileid & 15;    // 0..15
  const int lane = threadIdx.x;    // 0..31
  const int half = lane >> 4;      // 0/1: K sub-pair select
  const int l16  = lane & 15;

  const float* arow = pooled + (size_t)(mt * 16 + l16) * 256 + half * 2;
  const float* brow = Wm     + (size_t)(nt * 16 + l16) * 256 + half * 2;

  v8f acc = {};
#pragma unroll 8
  for (int k0 = 0; k0 < 256; k0 += 4) {
    v2f a, b;
    a.x = arow[k0];     a.y = arow[k0 + 1];
    b.x = brow[k0];     b.y = brow[k0 + 1];
    // (neg_a, A, neg_b, B, c_mod, C, reuse_a, reuse_b)
    acc = __builtin_amdgcn_wmma_f32_16x16x4_f32(false, a, false, b,
                                                (short)0, acc, false, false);
  }

  const int n  = nt * 16 + l16;
  const float bs = bias[n];
#pragma unroll
  for (int v = 0; v < 8; ++v) {
    const int m = mt * 16 + v + half * 8;
    const float val = acc[v] + bs;
    att[(size_t)m * 256 + n] = 1.0f / (1.0f + expf(-val));
  }
}

// ---------------------------------------------------------------------------
// Pass 3: out = x * att[b][c][h/128][w/128], streamed as float4.
// A 32-lane wave covers 128 consecutive floats = exactly one tile-width, so
// all lanes in a wave share one att value (single L2-friendly broadcast load).
// Non-temporal stores for the 2 GiB output stream.
// ---------------------------------------------------------------------------
__global__ __launch_bounds__(256) void apply_kernel(const float* __restrict__ x,
                                                    const float* __restrict__ att,
                                                    float* __restrict__ out) {
  const unsigned idx = blockIdx.x * 256u + threadIdx.x;  // float4 index < 2^27
  const unsigned e = idx << 2;                           // element index
  const unsigned w = e & 511u;
  const unsigned h = (e >> 9) & 511u;
  const unsigned c = (e >> 18) & 255u;
  const unsigned b = e >> 26;
  const unsigned t = (h >> 7) * 4u + (w >> 7);

  const float g = att[((size_t)(b * 16u + t)) * 256u + c];

  const v4f* xi = (const v4f*)x;
  v4f* oi = (v4f*)out;
  v4f v = xi[(size_t)idx];
  v *= g;
  __builtin_nontemporal_store(v, oi + (size_t)idx);
}

// ---------------------------------------------------------------------------
extern "C" void kernel_launch(void* const* d_in, const int* in_sizes, int n_in,
                              void* d_out, int out_size, void* d_ws, size_t ws_size,
                              hipStream_t stream) {
  const float* x    = (const float*)d_in[0];   // (8,256,512,512)
  const float* Wm   = (const float*)d_in[1];   // (256,256) row-major (o,c)
  const float* bias = (const float*)d_in[2];   // (256,)
  float* out = (float*)d_out;

  float* pooled = (float*)d_ws;                // 128*256 floats
  float* att    = pooled + 128 * 256;          // 128*256 floats

  // Pass 1: 8*256*16 = 32768 tile reductions
  pool_kernel<<<32768, 256, 0, stream>>>(x, pooled);
  // Pass 2: 128 16x16 WMMA output tiles, one wave32 each
  gate_kernel<<<128, 32, 0, stream>>>(pooled, Wm, bias, att);
  // Pass 3: 2^27 float4 elements / 256 threads
  apply_kernel<<<524288, 256, 0, stream>>>(x, att, out);
}